// Direction_80573586472978
// MI455X (gfx1250) — compile-verified
//
#include <hip/hip_runtime.h>

typedef __attribute__((ext_vector_type(2))) float v2f;
typedef __attribute__((ext_vector_type(8))) float v8f;
typedef int v4i_vs __attribute__((vector_size(16)));

#define B_ROWS   131072
#define N_DIM    512
#define K_DIM    512

#define TM   128   // block tile M
#define TN   64    // block tile N
#define KC   32    // K chunk staged in LDS
#define LDST 36    // LDS row stride in floats: rows 16B-aligned, conflict-friendly

#if defined(__has_builtin)
#if __has_builtin(__builtin_amdgcn_global_load_async_to_lds_b128)
#define HAS_ASYNC_LDS 1
#endif
#endif

// generic->AS1 / generic->AS3 conversion via integer round-trip, typed as the
// int4* pointers the async-to-lds builtin expects.
#define AS1_PTR(p) ((__attribute__((address_space(1))) v4i_vs*)(uintptr_t)(p))
#define AS3_PTR(p) ((__attribute__((address_space(3))) v4i_vs*)(uintptr_t)(p))

__device__ __forceinline__ void wait_async_0() {
#if defined(__has_builtin)
#if __has_builtin(__builtin_amdgcn_s_wait_asynccnt)
    __builtin_amdgcn_s_wait_asynccnt(0);
    return;
#else
    asm volatile("s_wait_asynccnt 0x0" ::: "memory");
    return;
#endif
#else
    asm volatile("s_wait_asynccnt 0x0" ::: "memory");
#endif
}

// ---------------------------------------------------------------------------
// Kernel 1: W = weight + 1e-8   (into workspace; QR will run in-place there)
// ---------------------------------------------------------------------------
__global__ void add_eps_kernel(const float* __restrict__ w,
                               float* __restrict__ o, int n) {
    int i = blockIdx.x * blockDim.x + threadIdx.x;
    if (i < n) o[i] = w[i] + 1e-8f;
}

// ---------------------------------------------------------------------------
// Kernel 2: in-place modified Gram-Schmidt QR of the 512x512 matrix in W.
// One workgroup of 512 threads (16 wave32s). After this, W holds Q with
// orthonormal columns (diag(R) > 0 convention).
// ---------------------------------------------------------------------------
__global__ __launch_bounds__(512) void qr_mgs_kernel(float* __restrict__ W) {
    __shared__ float qcol[K_DIM];
    __shared__ float red[K_DIM];
    const int t = threadIdx.x;

    for (int j = 0; j < N_DIM; ++j) {
        float v = W[(size_t)t * N_DIM + j];
        red[t] = v * v;
        __syncthreads();
        #pragma unroll
        for (int s = 256; s > 0; s >>= 1) {
            if (t < s) red[t] += red[t + s];
            __syncthreads();
        }
        const float inv = rsqrtf(red[0]);
        const float q = v * inv;
        W[(size_t)t * N_DIM + j] = q;
        qcol[t] = q;
        __syncthreads();

        const int c = j + 1 + t;
        if (c < N_DIM) {
            float dot = 0.f;
            for (int i = 0; i < K_DIM; ++i)
                dot += qcol[i] * W[(size_t)i * N_DIM + c];
            for (int i = 0; i < K_DIM; ++i)
                W[(size_t)i * N_DIM + c] -= dot * qcol[i];
        }
        __syncthreads();
    }
}

// ---------------------------------------------------------------------------
// WMMA compute on one staged K-chunk (KC=32) for a 32x32 wave tile.
// ---------------------------------------------------------------------------
__device__ __forceinline__ void mma_chunk(const float* __restrict__ Ab,
                                          const float* __restrict__ Bb,
                                          int wm, int wn, int fm, int kh,
                                          v8f acc[2][2]) {
    #pragma unroll
    for (int kk = 0; kk < KC; kk += 4) {
        v2f afrag[2], bfrag[2];
        #pragma unroll
        for (int mt = 0; mt < 2; ++mt)
            afrag[mt] = *(const v2f*)(Ab + (wm * 32 + mt * 16 + fm) * LDST +
                                      kk + kh);
        #pragma unroll
        for (int nt = 0; nt < 2; ++nt)
            bfrag[nt] = *(const v2f*)(Bb + (wn * 32 + nt * 16 + fm) * LDST +
                                      kk + kh);
        #pragma unroll
        for (int mt = 0; mt < 2; ++mt) {
            #pragma unroll
            for (int nt = 0; nt < 2; ++nt) {
                acc[mt][nt] = __builtin_amdgcn_wmma_f32_16x16x4_f32(
                    false, afrag[mt], false, bfrag[nt],
                    (short)0, acc[mt][nt], false, false);
            }
        }
    }
}

// ---------------------------------------------------------------------------
// Kernel 3: out[b,n] = sum_m input[b,m] * Q[n,m]   (FP32 WMMA GEMM)
//   A = input (B_ROWS x 512, row-major), B(k,n) = Qm[n*512 + k] (rows of Q).
// 256 threads = 8 waves (4x2 M-x-N wave grid), wave tile 32x32 = 2x2 frags.
// Double-buffered LDS; global->LDS staging via async-to-LDS when available.
// ---------------------------------------------------------------------------
__global__ __launch_bounds__(256) void direction_gemm_kernel(
    const float* __restrict__ A,
    const float* __restrict__ Qm,
    float* __restrict__ out) {
    __shared__ __align__(16) float As[2][TM * LDST];
    __shared__ __align__(16) float Bs[2][TN * LDST];

    const int tid  = threadIdx.x;
    const int lane = tid & 31;
    const int wave = tid >> 5;
    const int wm   = wave & 3;         // wave's M slot (32 rows each)
    const int wn   = wave >> 2;        // wave's N slot (32 cols each)
    const int fm   = lane & 15;        // M (for A) / N (for B,C) within frag
    const int kh   = (lane >> 4) * 2;  // K sub-offset per ISA A/B layout

    const int rowBase = blockIdx.x * TM;
    const int colBase = blockIdx.y * TN;

    v8f acc[2][2] = {};

#if defined(HAS_ASYNC_LDS)
    // ---- async global->LDS staging path (no VGPR staging, no ds_store) ----
    #define STAGE_CHUNK(BUFI, KCOFF)                                          \
        do {                                                                  \
            _Pragma("unroll")                                                 \
            for (int i = 0; i < 4; ++i) {                                     \
                int q = tid + 256 * i, r = q >> 3, s = q & 7;                 \
                __builtin_amdgcn_global_load_async_to_lds_b128(               \
                    AS1_PTR(A + (size_t)(rowBase + r) * K_DIM + (KCOFF) +     \
                            s * 4),                                           \
                    AS3_PTR(&As[(BUFI)][r * LDST + s * 4]), 0, 0);            \
            }                                                                 \
            _Pragma("unroll")                                                 \
            for (int i = 0; i < 2; ++i) {                                     \
                int q = tid + 256 * i, r = q >> 3, s = q & 7;                 \
                __builtin_amdgcn_global_load_async_to_lds_b128(               \
                    AS1_PTR(Qm + (size_t)(colBase + r) * K_DIM + (KCOFF) +    \
                            s * 4),                                           \
                    AS3_PTR(&Bs[(BUFI)][r * LDST + s * 4]), 0, 0);            \
            }                                                                 \
        } while (0)

    STAGE_CHUNK(0, 0);
    wait_async_0();
    __syncthreads();

    int buf = 0;
    for (int kc = 0; kc < K_DIM; kc += KC) {
        const bool more = (kc + KC) < K_DIM;
        if (more) STAGE_CHUNK(buf ^ 1, kc + KC);

        mma_chunk(&As[buf][0], &Bs[buf][0], wm, wn, fm, kh, acc);

        wait_async_0();
        __syncthreads();
        buf ^= 1;
    }
    #undef STAGE_CHUNK
#else
    // ---- fallback: synchronous loads staged through VGPRs ----
    float4 aReg[4];
    float4 bReg[2];
    #pragma unroll
    for (int i = 0; i < 4; ++i) {
        int q = tid + 256 * i, r = q >> 3, s = q & 7;
        aReg[i] = *(const float4*)(A + (size_t)(rowBase + r) * K_DIM + s * 4);
    }
    #pragma unroll
    for (int i = 0; i < 2; ++i) {
        int q = tid + 256 * i, r = q >> 3, s = q & 7;
        bReg[i] = *(const float4*)(Qm + (size_t)(colBase + r) * K_DIM + s * 4);
    }
    #pragma unroll
    for (int i = 0; i < 4; ++i) {
        int q = tid + 256 * i, r = q >> 3, s = q & 7;
        *(float4*)&As[0][r * LDST + s * 4] = aReg[i];
    }
    #pragma unroll
    for (int i = 0; i < 2; ++i) {
        int q = tid + 256 * i, r = q >> 3, s = q & 7;
        *(float4*)&Bs[0][r * LDST + s * 4] = bReg[i];
    }
    __syncthreads();

    int buf = 0;
    for (int kc = 0; kc < K_DIM; kc += KC) {
        const bool more = (kc + KC) < K_DIM;
        if (more) {
            #pragma unroll
            for (int i = 0; i < 4; ++i) {
                int q = tid + 256 * i, r = q >> 3, s = q & 7;
                aReg[i] = *(const float4*)(A + (size_t)(rowBase + r) * K_DIM +
                                           (kc + KC) + s * 4);
            }
            #pragma unroll
            for (int i = 0; i < 2; ++i) {
                int q = tid + 256 * i, r = q >> 3, s = q & 7;
                bReg[i] = *(const float4*)(Qm + (size_t)(colBase + r) * K_DIM +
                                           (kc + KC) + s * 4);
            }
        }

        mma_chunk(&As[buf][0], &Bs[buf][0], wm, wn, fm, kh, acc);

        __syncthreads();
        if (more) {
            const int nb = buf ^ 1;
            #pragma unroll
            for (int i = 0; i < 4; ++i) {
                int q = tid + 256 * i, r = q >> 3, s = q & 7;
                *(float4*)&As[nb][r * LDST + s * 4] = aReg[i];
            }
            #pragma unroll
            for (int i = 0; i < 2; ++i) {
                int q = tid + 256 * i, r = q >> 3, s = q & 7;
                *(float4*)&Bs[nb][r * LDST + s * 4] = bReg[i];
            }
            __syncthreads();
            buf = nb;
        }
    }
#endif

    // ---- epilogue: write C frags (lane: N = lane&15, M = (lane>>4)*8 + v) ----
    const int mrow = rowBase + wm * 32 + (lane >> 4) * 8;
    const int ncol = colBase + wn * 32 + fm;
    #pragma unroll
    for (int mt = 0; mt < 2; ++mt) {
        #pragma unroll
        for (int nt = 0; nt < 2; ++nt) {
            #pragma unroll
            for (int v = 0; v < 8; ++v) {
                out[(size_t)(mrow + mt * 16 + v) * N_DIM + ncol + nt * 16] =
                    acc[mt][nt][v];
            }
        }
    }
}

// ---------------------------------------------------------------------------
extern "C" void kernel_launch(void* const* d_in, const int* in_sizes, int n_in,
                              void* d_out, int out_size, void* d_ws, size_t ws_size,
                              hipStream_t stream) {
    const float* inp = (const float*)d_in[0];   // (131072, 512) f32
    const float* wgt = (const float*)d_in[1];   // (512, 512) f32
    float* out = (float*)d_out;                 // (131072, 512) f32
    float* Q   = (float*)d_ws;                  // 512*512 f32 workspace

    const int n_w = N_DIM * K_DIM;
    add_eps_kernel<<<(n_w + 255) / 256, 256, 0, stream>>>(wgt, Q, n_w);
    qr_mgs_kernel<<<1, 512, 0, stream>>>(Q);

    dim3 grid(B_ROWS / TM, N_DIM / TN);  // 1024 x 8 blocks
    direction_gemm_kernel<<<grid, 256, 0, stream>>>(inp, Q, out);
}